// IsoGMMLoss_7834020348325
// MI455X (gfx1250) — compile-verified
//
#include <hip/hip_runtime.h>

typedef __attribute__((ext_vector_type(2))) float v2f;
typedef __attribute__((ext_vector_type(8))) float v8f;
typedef __attribute__((ext_vector_type(4))) unsigned int v4u;
typedef __attribute__((ext_vector_type(8))) int v8i;
typedef __attribute__((ext_vector_type(4))) int v4i;

#define N_TOT 131072
#define K_DIM 64
#define D_DIM 128
#define GEMM_BLOCKS 256
#define TILE_N 64                         // rows staged per TDM tile
#define NTILES ((N_TOT / GEMM_BLOCKS) / TILE_N)   // 8 tiles per block
#define STAT_THREADS 256
#define STAT_BLOCKS (N_TOT / STAT_THREADS)

// d_ws layout (floats): [0] = S1, [64..127] = rcolsum c[k], [128..8319] = M = r^T X (K x D)
#define WS_S1 0
#define WS_RCOL 64
#define WS_M 128
#define WS_FLOATS (WS_M + K_DIM * D_DIM)

// -------------------------------------------------------------------------
// TDM descriptor build + issue: 2D tile (rows x width f32, contiguous rows)
// from global memory into LDS. Bitfields per CDNA5 ISA 8.3/8.4 (D# groups).
// This toolchain exposes the 6-arg builtin:
//   (uint32x4 g0, int32x8 g1, int32x4 g2, int32x4 g3, int32x8 g1b, i32 cpol)
// -------------------------------------------------------------------------
__device__ __forceinline__ void tdm_load_2d(unsigned lds_off, const float* gptr,
                                            unsigned width, unsigned rows) {
    unsigned long long ga = (unsigned long long)(const void*)gptr;
    v4u g0;
    g0.x = 1u;                                           // count=1 (valid), user mode
    g0.y = lds_off;                                      // lds_addr (bytes)
    g0.z = (unsigned)(ga & 0xFFFFFFFFu);                 // global_addr[31:0]
    g0.w = (unsigned)((ga >> 32) & 0x01FFFFFFu) | (2u << 30);  // addr[56:32] | type=2

    v8i g1;
    g1[0] = (int)(2u << 16);                 // workgroup_mask=0, data_size=2 (4B)
    g1[1] = (int)((width & 0xFFFFu) << 16);  // tensor_dim0[15:0]
    g1[2] = (int)((rows  & 0xFFFFu) << 16);  // tensor_dim0[31:16]=0 | tensor_dim1[15:0]
    g1[3] = (int)((width & 0xFFFFu) << 16);  // tensor_dim1[31:16]=0 | tile_dim0
    g1[4] = (int)(rows & 0xFFFFu);           // tile_dim1 | tile_dim2=0
    g1[5] = (int)width;                      // tensor_dim0_stride[31:0]
    g1[6] = 0;                               // stride[47:32]=0 | tensor_dim1_stride lo
    g1[7] = 0;

    v4i gz4 = (v4i){0, 0, 0, 0};             // groups 2/3 unused (2D tile)
    v8i gz8 = (v8i){0, 0, 0, 0, 0, 0, 0, 0}; // unused second descriptor slot
    __builtin_amdgcn_tensor_load_to_lds(g0, g1, gz4, gz4, gz8, 0);
}

// -------------------------------------------------------------------------
// Kernel 1: M = r^T @ X (64 x 128, accumulated over N) via V_WMMA_F32_16X16X4_F32,
// fed from LDS tiles staged by the Tensor Data Mover (double-buffered, wave 0
// issues, TENSORcnt-gated, pipelined one tile ahead).
// Wave w owns d-columns [16w,16w+16) and all 4 k-tiles of the 64x128 output.
// -------------------------------------------------------------------------
__global__ void __launch_bounds__(256)
iso_gmm_rtx_gemm(const float* __restrict__ X, const float* __restrict__ r,
                 float* __restrict__ ws) {
    __shared__ float Xs[2][TILE_N * D_DIM];   // 2 x 32 KB
    __shared__ float Rs[2][TILE_N * K_DIM];   // 2 x 16 KB

    // Never-executed store so the compiler cannot treat the DMA-written LDS
    // as unwritten and fold the loads.
    if (ws == (float*)1) { Xs[0][threadIdx.x] = 0.f; Rs[0][threadIdx.x] = 0.f; }

    const int lane = threadIdx.x & 31;
    const int wave = threadIdx.x >> 5;
    const int half = lane >> 4;
    const int l    = lane & 15;
    const int d0   = wave * 16;

    const size_t nbase = (size_t)blockIdx.x * (N_TOT / GEMM_BLOCKS);
    const unsigned xs_base = (unsigned)(unsigned long long)&Xs[0][0];
    const unsigned rs_base = (unsigned)(unsigned long long)&Rs[0][0];

    v8f acc[4];
    acc[0] = (v8f){}; acc[1] = (v8f){}; acc[2] = (v8f){}; acc[3] = (v8f){};

    // Prologue: stage tile 0.
    if (threadIdx.x < 32) {
        tdm_load_2d(xs_base, X + nbase * D_DIM, D_DIM, TILE_N);
        tdm_load_2d(rs_base, r + nbase * K_DIM, K_DIM, TILE_N);
    }

    for (int t = 0; t < NTILES; ++t) {
        if (threadIdx.x < 32) {
            if (t + 1 < NTILES) {
                const size_t nn = nbase + (size_t)(t + 1) * TILE_N;
                const unsigned buf = (unsigned)((t + 1) & 1);
                tdm_load_2d(xs_base + buf * (TILE_N * D_DIM * 4), X + nn * D_DIM, D_DIM, TILE_N);
                tdm_load_2d(rs_base + buf * (TILE_N * K_DIM * 4), r + nn * K_DIM, K_DIM, TILE_N);
                __builtin_amdgcn_s_wait_tensorcnt(2);   // tile t's pair complete
            } else {
                __builtin_amdgcn_s_wait_tensorcnt(0);
            }
        }
        __syncthreads();                                 // tile t visible to all waves

        const float* xb = Xs[t & 1];
        const float* rb = Rs[t & 1];
        #pragma unroll 4
        for (int n = 0; n < TILE_N; n += 4) {
            const int row = n + 2 * half;
            v2f b;
            b.x = xb[row * D_DIM + d0 + l];
            b.y = xb[(row + 1) * D_DIM + d0 + l];
            const float* rr = rb + row * K_DIM + l;
            v2f a0; a0.x = rr[0];  a0.y = rr[K_DIM];          // ds_load_2addr pairs
            v2f a1; a1.x = rr[16]; a1.y = rr[16 + K_DIM];
            v2f a2; a2.x = rr[32]; a2.y = rr[32 + K_DIM];
            v2f a3; a3.x = rr[48]; a3.y = rr[48 + K_DIM];
            acc[0] = __builtin_amdgcn_wmma_f32_16x16x4_f32(false, a0, false, b,
                                                           (short)0, acc[0], false, false);
            acc[1] = __builtin_amdgcn_wmma_f32_16x16x4_f32(false, a1, false, b,
                                                           (short)0, acc[1], false, false);
            acc[2] = __builtin_amdgcn_wmma_f32_16x16x4_f32(false, a2, false, b,
                                                           (short)0, acc[2], false, false);
            acc[3] = __builtin_amdgcn_wmma_f32_16x16x4_f32(false, a3, false, b,
                                                           (short)0, acc[3], false, false);
        }
        __syncthreads();                                 // done reading buf before DMA reuse
    }

    // C/D layout: VGPR v, lanes 0-15 -> row v, lanes 16-31 -> row v+8; col = l.
    float* M = ws + WS_M;
    const int d = d0 + l;
    #pragma unroll
    for (int v = 0; v < 8; ++v) {
        const int krow = v + 8 * half;
        unsafeAtomicAdd(&M[(0 * 16 + krow) * D_DIM + d], acc[0][v]);
        unsafeAtomicAdd(&M[(1 * 16 + krow) * D_DIM + d], acc[1][v]);
        unsafeAtomicAdd(&M[(2 * 16 + krow) * D_DIM + d], acc[2][v]);
        unsafeAtomicAdd(&M[(3 * 16 + krow) * D_DIM + d], acc[3][v]);
    }
}

// -------------------------------------------------------------------------
// Kernel 2: per-row stats. One thread per n.
//   S1 += ||X[n]||^2 * sum_k r[n,k];   c[k] += r[n,k]
// X+r (96 MB) fit in the 192 MB L2 after kernel 1 -> mostly L2 hits here.
// -------------------------------------------------------------------------
__global__ void __launch_bounds__(STAT_THREADS)
iso_gmm_row_stats(const float* __restrict__ X, const float* __restrict__ r,
                  float* __restrict__ ws) {
    __shared__ float rcol[K_DIM];
    __shared__ float red[STAT_THREADS];
    const int tid = threadIdx.x;
    if (tid < K_DIM) rcol[tid] = 0.0f;
    __syncthreads();

    const size_t n = (size_t)blockIdx.x * STAT_THREADS + tid;
    const float4* xr = (const float4*)(X + n * D_DIM);
    const float4* rr = (const float4*)(r + n * K_DIM);

    float xsq = 0.0f;
    #pragma unroll
    for (int j = 0; j < D_DIM / 4; ++j) {
        float4 v = xr[j];
        xsq += v.x * v.x + v.y * v.y + v.z * v.z + v.w * v.w;
    }

    float rs = 0.0f;
    #pragma unroll
    for (int j = 0; j < K_DIM / 4; ++j) {
        float4 v = rr[j];
        rs += v.x + v.y + v.z + v.w;
        atomicAdd(&rcol[4 * j + 0], v.x);   // ds_add_f32
        atomicAdd(&rcol[4 * j + 1], v.y);
        atomicAdd(&rcol[4 * j + 2], v.z);
        atomicAdd(&rcol[4 * j + 3], v.w);
    }

    red[tid] = xsq * rs;
    __syncthreads();
    for (int s = STAT_THREADS / 2; s > 0; s >>= 1) {
        if (tid < s) red[tid] += red[tid + s];
        __syncthreads();
    }
    if (tid == 0)    unsafeAtomicAdd(&ws[WS_S1], red[0]);
    if (tid < K_DIM) unsafeAtomicAdd(&ws[WS_RCOL + tid], rcol[tid]);
}

// -------------------------------------------------------------------------
// Kernel 3: finalize. loss = (S1 + sum_{k,d} mu^2*c[k] - 2*mu*M) / (N*K)
// -------------------------------------------------------------------------
__global__ void __launch_bounds__(256)
iso_gmm_finalize(const float* __restrict__ mus, const float* __restrict__ ws,
                 float* __restrict__ out) {
    __shared__ float red[256];
    const int tid = threadIdx.x;
    const float* c = ws + WS_RCOL;
    const float* M = ws + WS_M;

    float local = 0.0f;
    for (int idx = tid; idx < K_DIM * D_DIM; idx += 256) {
        const int k = idx >> 7;
        const float m = mus[idx];
        local += m * (m * c[k] - 2.0f * M[idx]);
    }
    red[tid] = local;
    __syncthreads();
    for (int s = 128; s > 0; s >>= 1) {
        if (tid < s) red[tid] += red[tid + s];
        __syncthreads();
    }
    if (tid == 0)
        out[0] = (red[0] + ws[WS_S1]) / ((float)N_TOT * (float)K_DIM);
}

extern "C" void kernel_launch(void* const* d_in, const int* in_sizes, int n_in,
                              void* d_out, int out_size, void* d_ws, size_t ws_size,
                              hipStream_t stream) {
    (void)in_sizes; (void)n_in; (void)out_size; (void)ws_size;
    const float* X   = (const float*)d_in[0];
    const float* r   = (const float*)d_in[1];
    const float* mus = (const float*)d_in[2];
    float* ws  = (float*)d_ws;
    float* out = (float*)d_out;

    (void)hipMemsetAsync(ws, 0, WS_FLOATS * sizeof(float), stream);

    iso_gmm_rtx_gemm<<<GEMM_BLOCKS, 256, 0, stream>>>(X, r, ws);
    iso_gmm_row_stats<<<STAT_BLOCKS, STAT_THREADS, 0, stream>>>(X, r, ws);
    iso_gmm_finalize<<<1, 256, 0, stream>>>(mus, ws, out);
}